// GraphConvolutionNeuralNetwork_31550829756770
// MI455X (gfx1250) — compile-verified
//
#include <hip/hip_runtime.h>

typedef __attribute__((ext_vector_type(2))) float v2f;
typedef __attribute__((ext_vector_type(8))) float v8f;

#define XS_STRIDE 132  // 128 + 4 pad: 16B-aligned rows, lanes land on distinct LDS banks

// ---------------- init: zero out1, deg = 1.0 (self loop) ----------------
__global__ __launch_bounds__(256) void k_init(float* __restrict__ out1,
                                              float* __restrict__ deg, int n) {
  int i = blockIdx.x * blockDim.x + threadIdx.x;
  int q = n * 64;  // n*256/4 float4s
  if (i < q) {
    float4 z = make_float4(0.f, 0.f, 0.f, 0.f);
    reinterpret_cast<float4*>(out1)[i] = z;
  }
  if (i < n) deg[i] = 1.0f;
}

// ---------------- degree accumulation over edges ----------------
__global__ __launch_bounds__(256) void k_degree(const long long* __restrict__ ei,
                                                float* __restrict__ deg, int E) {
  int e = blockIdx.x * blockDim.x + threadIdx.x;
  if (e < E) {
    long long d = ei[(long long)E + e];
    __hip_atomic_fetch_add(&deg[d], 1.0f, __ATOMIC_RELAXED, __HIP_MEMORY_SCOPE_AGENT);
  }
}

// ---------------- dis = rsqrt(deg) in place ----------------
__global__ __launch_bounds__(256) void k_rsqrt(float* __restrict__ deg, int n) {
  int i = blockIdx.x * blockDim.x + threadIdx.x;
  if (i < n) {
    float d = deg[i];
    deg[i] = (d > 0.f) ? rsqrtf(d) : 0.f;
  }
}

// ---------------- H = X @ W1  (fp32 WMMA 16x16x4) ----------------
// Block: 512 threads = 16 waves. Block owns one 16-row M-tile of X (staged in LDS),
// wave w owns N-tile w (16 of 256 output cols). K=128 unrolled -> 32 v_wmma per wave.
__global__ __launch_bounds__(512) void k_gemm(const float* __restrict__ X,
                                              const float* __restrict__ W1,
                                              float* __restrict__ H, int n) {
  __shared__ float xs[16 * XS_STRIDE];
  const int tid = threadIdx.x;
  const int m0 = blockIdx.x * 16;

  // Stage 16x128 X tile: 2048 floats / 512 threads = one float4 each (coalesced).
  {
    int idx = tid * 4;
    int row = idx >> 7;        // 0..15
    int col = idx & 127;       // multiple of 4
    int grow = m0 + row;
    if (grow >= n) grow = n - 1;  // defensive clamp (n is a multiple of 16 here)
    float4 v = *reinterpret_cast<const float4*>(X + (size_t)grow * 128 + col);
    *reinterpret_cast<float4*>(&xs[row * XS_STRIDE + col]) = v;
  }
  __syncthreads();

  const int wave = tid >> 5;       // 0..15 -> N tile
  const int lane = tid & 31;
  const int lo = lane & 15;        // M row (A) / N col (B,C,D) within tile
  const int hi = lane >> 4;        // K half-select
  const int col = wave * 16 + lo;  // global output column (0..255)

  v8f c = {0.f, 0.f, 0.f, 0.f, 0.f, 0.f, 0.f, 0.f};

#pragma unroll
  for (int k0 = 0; k0 < 128; k0 += 4) {
    const int k = k0 + 2 * hi;  // VGPR0 holds K=k, VGPR1 holds K=k+1 (per 16x4 f32 A layout)
    v2f a = *reinterpret_cast<const v2f*>(&xs[lo * XS_STRIDE + k]);  // 8B-aligned LDS read
    v2f b;
    b.x = W1[k * 256 + col];
    b.y = W1[(k + 1) * 256 + col];
    c = __builtin_amdgcn_wmma_f32_16x16x4_f32(false, a, false, b, (short)0, c,
                                              false, false);
  }

#pragma unroll
  for (int r = 0; r < 8; ++r) {
    int row = m0 + r + hi * 8;  // C/D layout: lanes 16-31 hold rows M=8..15
    if (row < n) H[(size_t)row * 256 + col] = c[r];
  }
}

// ---------------- layer-1 scatter: out1[dst] += norm * H[src] ----------------
// One wave per edge; 8 features/lane, stride-32 (coalesced loads + L2 atomics).
__global__ __launch_bounds__(256) void k_agg1(const long long* __restrict__ ei,
                                              const float* __restrict__ H,
                                              const float* __restrict__ dis,
                                              float* __restrict__ out1, int E) {
  int e = blockIdx.x * 8 + (threadIdx.x >> 5);
  if (e >= E) return;
  int lane = threadIdx.x & 31;
  long long s = ei[e];
  long long d = ei[(long long)E + e];
  float nrm = dis[s] * dis[d];
  const float* hrow = H + (size_t)s * 256;
  float* orow = out1 + (size_t)d * 256;
#pragma unroll
  for (int j = 0; j < 8; ++j) {
    int cidx = lane + 32 * j;
    __hip_atomic_fetch_add(&orow[cidx], hrow[cidx] * nrm, __ATOMIC_RELAXED,
                           __HIP_MEMORY_SCOPE_AGENT);
  }
}

// ---------------- finalize L1 (+self loop, +b1, ReLU) and L2 node transform ----------------
// One wave per node: h2[i] = relu(out1_i + dis_i^2*H_i + b1) . W2 ; seed d_out with b2 + self term.
__global__ __launch_bounds__(256) void k_final(const float* __restrict__ out1,
                                               const float* __restrict__ H,
                                               const float* __restrict__ dis,
                                               const float* __restrict__ b1,
                                               const float* __restrict__ W2,
                                               const float* __restrict__ b2,
                                               float* __restrict__ h2,
                                               float* __restrict__ out2, int n) {
  int i = blockIdx.x * 8 + (threadIdx.x >> 5);
  if (i >= n) return;
  int lane = threadIdx.x & 31;
  float sn = dis[i] * dis[i];
  float dot = 0.f;
#pragma unroll
  for (int j = 0; j < 8; ++j) {
    int c = lane + 32 * j;
    float v = out1[(size_t)i * 256 + c] + H[(size_t)i * 256 + c] * sn + b1[c];
    v = fmaxf(v, 0.f);
    dot = fmaf(v, W2[c], dot);
  }
  // wave32 butterfly reduction
#pragma unroll
  for (int off = 16; off >= 1; off >>= 1) dot += __shfl_xor(dot, off, 32);
  if (lane == 0) {
    h2[i] = dot;
    out2[i] = b2[0] + dot * sn;  // bias + self-loop message
  }
}

// ---------------- layer-2 scatter: out2[dst] += norm * h2[src] ----------------
__global__ __launch_bounds__(256) void k_agg2(const long long* __restrict__ ei,
                                              const float* __restrict__ h2,
                                              const float* __restrict__ dis,
                                              float* __restrict__ out2, int E) {
  int e = blockIdx.x * blockDim.x + threadIdx.x;
  if (e < E) {
    long long s = ei[e];
    long long d = ei[(long long)E + e];
    float v = h2[s] * dis[s] * dis[d];
    __hip_atomic_fetch_add(&out2[d], v, __ATOMIC_RELAXED, __HIP_MEMORY_SCOPE_AGENT);
  }
}

extern "C" void kernel_launch(void* const* d_in, const int* in_sizes, int n_in,
                              void* d_out, int out_size, void* d_ws, size_t ws_size,
                              hipStream_t stream) {
  const float* X = (const float*)d_in[0];
  const long long* ei = (const long long*)d_in[1];  // int64 edge_index [2, E]
  const float* W1 = (const float*)d_in[2];
  const float* b1 = (const float*)d_in[3];
  const float* W2 = (const float*)d_in[4];
  const float* b2 = (const float*)d_in[5];
  float* out2 = (float*)d_out;

  const int n = in_sizes[0] / 128;  // 50000
  const int E = in_sizes[1] / 2;    // 1600000

  // Workspace layout (floats): H[n*256] | out1[n*256] | dis[n] | h2[n]  (~103 MB)
  float* ws = (float*)d_ws;
  float* H = ws;
  float* out1 = H + (size_t)n * 256;
  float* dis = out1 + (size_t)n * 256;
  float* h2 = dis + n;

  int initGrid = (n * 64 + 255) / 256;
  k_init<<<initGrid, 256, 0, stream>>>(out1, dis, n);
  k_degree<<<(E + 255) / 256, 256, 0, stream>>>(ei, dis, E);
  k_rsqrt<<<(n + 255) / 256, 256, 0, stream>>>(dis, n);
  k_gemm<<<(n + 15) / 16, 512, 0, stream>>>(X, W1, H, n);
  k_agg1<<<(E + 7) / 8, 256, 0, stream>>>(ei, H, dis, out1, E);
  k_final<<<(n + 7) / 8, 256, 0, stream>>>(out1, H, dis, b1, W2, b2, h2, out2, n);
  k_agg2<<<(E + 255) / 256, 256, 0, stream>>>(ei, h2, dis, out2, E);
}